// MultiGCN_66812511257314
// MI455X (gfx1250) — compile-verified
//
#include <hip/hip_runtime.h>

typedef float v2f __attribute__((ext_vector_type(2)));
typedef float v8f __attribute__((ext_vector_type(8)));

#define D_IN 128   // K dimension is 128 for all four layers

// ---------------- degree / norm kernels ----------------
__global__ void k_dinv_init(float* dinv, int n) {
    int i = blockIdx.x * blockDim.x + threadIdx.x;
    if (i < n) dinv[i] = 1.0f;  // self loop contributes 1
}

__global__ void k_deg_count(const int* __restrict__ dst, float* dinv, int E) {
    int e = blockIdx.x * blockDim.x + threadIdx.x;
    if (e < E) atomicAdd(&dinv[dst[e]], 1.0f);
}

__global__ void k_dinv_finalize(float* dinv, int n) {
    int i = blockIdx.x * blockDim.x + threadIdx.x;
    if (i < n) dinv[i] = rsqrtf(dinv[i]);
}

// ---------------- WMMA GEMM: H = X @ W  (K = 128 fixed, OUT compile-time) ----
// grid.x = nrows/16, blockDim = 32 * (OUT/16). One wave per 16x16 C tile.
// OUT templated so B loads / C stores use immediate offsets (fit 24-bit IOFFSET).
template <int OUT>
__global__ __launch_bounds__(256) void k_gemm_wmma(
    const float* __restrict__ X, const float* __restrict__ W,
    float* __restrict__ H) {
    __shared__ float As[16 * 132];               // padded stride: 132 = 4 mod 64 banks
    const int tileRow = blockIdx.x * 16;
    const int tid = threadIdx.x;

    // cooperative load of A tile (16 x 128 f32) into LDS, float4 coalesced
    for (int i = tid * 4; i < 16 * 128; i += blockDim.x * 4) {
        const int r = i >> 7, c = i & 127;
        float4 v = *(const float4*)&X[(size_t)(tileRow + r) * D_IN + c];
        As[r * 132 + c + 0] = v.x;
        As[r * 132 + c + 1] = v.y;
        As[r * 132 + c + 2] = v.z;
        As[r * 132 + c + 3] = v.w;
    }
    __syncthreads();

    const int lane    = tid & 31;
    const int wave    = tid >> 5;
    const int colBase = wave * 16;
    const int mrow    = lane & 15;        // A row / B col handled by this lane
    const int khalf   = (lane >> 4) * 2;  // 0 for lanes 0-15, 2 for lanes 16-31

    const float* __restrict__ Wbase = W + (size_t)khalf * OUT + colBase + mrow;
    const float* __restrict__ Abase = As + mrow * 132 + khalf;

    v8f c = {};
    #pragma unroll
    for (int kk = 0; kk < D_IN; kk += 4) {
        v2f a, b;
        a.x = Abase[kk + 0];                  // LDS, immediate offsets
        a.y = Abase[kk + 1];
        b.x = Wbase[(kk + 0) * OUT];          // global, immediate offsets
        b.y = Wbase[(kk + 1) * OUT];
        c = __builtin_amdgcn_wmma_f32_16x16x4_f32(
                /*neg_a=*/false, a, /*neg_b=*/false, b,
                /*c_mod=*/(short)0, c, /*reuse_a=*/false, /*reuse_b=*/false);
    }

    // C layout: VGPR v -> row (v + 8*(lane>=16)), col = lane&15
    const int half = (lane >> 4) << 3;
    float* __restrict__ Hbase = H + (size_t)(tileRow + half) * OUT + colBase + mrow;
    #pragma unroll
    for (int v = 0; v < 8; ++v) {
        Hbase[v * OUT] = c[v];                // immediate offsets
    }
}

// ---------------- self-loop init: agg = h * dinv[n]^2 ----------------
// dimShift = log2(dim): 7 for 128, 6 for 64
__global__ void k_self_init(const float* __restrict__ H, const float* __restrict__ dinv,
                            float* __restrict__ agg, long long total, int dimShift) {
    size_t i4 = ((size_t)blockIdx.x * blockDim.x + threadIdx.x) * 4;
    if (i4 >= (size_t)total) return;
    int node = (int)(i4 >> dimShift);
    float dv = dinv[node];
    float s = dv * dv;
    float4 h = *(const float4*)&H[i4];
    float4 o = make_float4(h.x * s, h.y * s, h.z * s, h.w * s);
    *(float4*)&agg[i4] = o;
}

// ---------------- edge scatter: agg[dst] += h[src] * dinv[s]*dinv[d] ----------------
// shift = log2(dim/4): 5 for dim=128, 4 for dim=64. One float4 per thread.
__global__ void k_edge_scatter(const int* __restrict__ src, const int* __restrict__ dst,
                               const float* __restrict__ dinv, const float* __restrict__ H,
                               float* __restrict__ agg, int E, int dim, int shift) {
    long long gid = (long long)blockIdx.x * blockDim.x + threadIdx.x;
    long long total = (long long)E << shift;
    if (gid >= total) return;
    int e = (int)(gid >> shift);
    int f = (int)(gid & ((1 << shift) - 1)) * 4;
    int s = src[e], d = dst[e];
    float nrm = dinv[s] * dinv[d];
    float4 hv = *(const float4*)&H[(size_t)s * dim + f];
    float* outp = &agg[(size_t)d * dim + f];
    atomicAdd(outp + 0, hv.x * nrm);
    atomicAdd(outp + 1, hv.y * nrm);
    atomicAdd(outp + 2, hv.z * nrm);
    atomicAdd(outp + 3, hv.w * nrm);
}

// ---------------- epilogue: out = [relu](agg + b) ----------------
__global__ void k_epilogue(const float* __restrict__ agg, const float* __restrict__ b,
                           float* __restrict__ out, long long total, int dimMask, int doRelu) {
    size_t i4 = ((size_t)blockIdx.x * blockDim.x + threadIdx.x) * 4;
    if (i4 >= (size_t)total) return;
    int f = (int)(i4 & (size_t)dimMask);
    float4 a = *(const float4*)&agg[i4];
    float4 o = make_float4(a.x + b[f], a.y + b[f + 1], a.z + b[f + 2], a.w + b[f + 3]);
    if (doRelu) {
        o.x = fmaxf(o.x, 0.0f); o.y = fmaxf(o.y, 0.0f);
        o.z = fmaxf(o.z, 0.0f); o.w = fmaxf(o.w, 0.0f);
    }
    *(float4*)&out[i4] = o;
}

extern "C" void kernel_launch(void* const* d_in, const int* in_sizes, int n_in,
                              void* d_out, int out_size, void* d_ws, size_t ws_size,
                              hipStream_t stream) {
    const float* x   = (const float*)d_in[0];
    const int*   ei  = (const int*)d_in[1];
    const float* W0  = (const float*)d_in[2];
    const float* b0  = (const float*)d_in[3];
    const float* W1  = (const float*)d_in[4];
    const float* b1  = (const float*)d_in[5];
    const float* W2  = (const float*)d_in[6];
    const float* b2  = (const float*)d_in[7];
    const float* W3  = (const float*)d_in[8];
    const float* b3  = (const float*)d_in[9];

    const int N = in_sizes[0] / D_IN;        // 100000
    const int E = in_sizes[1] / 2;           // 1600000
    const int H = 128, O = 64;

    const int* src = ei;                     // edge_index[0]
    const int* dst = ei + E;                 // edge_index[1]

    // workspace carve-up (floats): dinv | bufA | bufB | bufC
    float* dinv = (float*)d_ws;
    float* bufA = dinv + N;
    float* bufB = bufA + (size_t)N * H;
    float* bufC = bufB + (size_t)N * H;

    const int T = 256;
    auto cdiv = [](long long a, long long b) { return (int)((a + b - 1) / b); };

    // ---- degree -> dinv ----
    k_dinv_init<<<cdiv(N, T), T, 0, stream>>>(dinv, N);
    k_deg_count<<<cdiv(E, T), T, 0, stream>>>(dst, dinv, E);
    k_dinv_finalize<<<cdiv(N, T), T, 0, stream>>>(dinv, N);

    const int nTiles = N / 16;               // 100000 % 16 == 0
    const long long elems128 = (long long)N * H;
    const long long elems64  = (long long)N * O;

    // ---- layer 0: x -> bufC (X1) ----
    k_gemm_wmma<128><<<nTiles, 256, 0, stream>>>(x, W0, bufB);
    k_self_init<<<cdiv(elems128 / 4, T), T, 0, stream>>>(bufB, dinv, bufC, elems128, 7);
    k_edge_scatter<<<cdiv((long long)E << 5, T), T, 0, stream>>>(src, dst, dinv, bufB, bufC, E, H, 5);
    k_epilogue<<<cdiv(elems128 / 4, T), T, 0, stream>>>(bufC, b0, bufC, elems128, 127, 1);

    // ---- layer 1: bufC -> bufA (X2) ----
    k_gemm_wmma<128><<<nTiles, 256, 0, stream>>>(bufC, W1, bufB);
    k_self_init<<<cdiv(elems128 / 4, T), T, 0, stream>>>(bufB, dinv, bufA, elems128, 7);
    k_edge_scatter<<<cdiv((long long)E << 5, T), T, 0, stream>>>(src, dst, dinv, bufB, bufA, E, H, 5);
    k_epilogue<<<cdiv(elems128 / 4, T), T, 0, stream>>>(bufA, b1, bufA, elems128, 127, 1);

    // ---- layer 2: bufA -> bufC (X3) ----
    k_gemm_wmma<128><<<nTiles, 256, 0, stream>>>(bufA, W2, bufB);
    k_self_init<<<cdiv(elems128 / 4, T), T, 0, stream>>>(bufB, dinv, bufC, elems128, 7);
    k_edge_scatter<<<cdiv((long long)E << 5, T), T, 0, stream>>>(src, dst, dinv, bufB, bufC, E, H, 5);
    k_epilogue<<<cdiv(elems128 / 4, T), T, 0, stream>>>(bufC, b2, bufC, elems128, 127, 1);

    // ---- layer 3 (output, dim 64, no relu): bufC -> d_out ----
    k_gemm_wmma<64><<<nTiles, 128, 0, stream>>>(bufC, W3, bufB);
    k_self_init<<<cdiv(elems64 / 4, T), T, 0, stream>>>(bufB, dinv, bufA, elems64, 6);
    k_edge_scatter<<<cdiv((long long)E << 4, T), T, 0, stream>>>(src, dst, dinv, bufB, bufA, E, O, 4);
    k_epilogue<<<cdiv(elems64 / 4, T), T, 0, stream>>>(bufA, b3, (float*)d_out, elems64, 63, 0);
}